// LatentLookup_88029649699282
// MI455X (gfx1250) — compile-verified
//
#include <hip/hip_runtime.h>
#include <math.h>

typedef __attribute__((ext_vector_type(2))) float v2f;
typedef __attribute__((ext_vector_type(8))) float v8f;
typedef int v4i_vec __attribute__((vector_size(16)));   // matches builtin's pointee type

#define SLICE_POINTS 4096   // 32 KB LDS staging tile of float2 points

// ---------------------------------------------------------------------------
// CDNA5 async global->LDS copy (ASYNCcnt path) with inline-asm fallback.
// Builtin signature (from hipcc diagnostic): (AS1 v4i*, AS3 v4i*, int, int).
// ---------------------------------------------------------------------------
__device__ __forceinline__ void async_copy_b128(const float* g, float* l) {
#if __has_builtin(__builtin_amdgcn_global_load_async_to_lds_b128)
  typedef v4i_vec __attribute__((address_space(1)))* gp_t;
  typedef v4i_vec __attribute__((address_space(3)))* lp_t;
  __builtin_amdgcn_global_load_async_to_lds_b128(
      (gp_t)(size_t)(const void*)g,
      (lp_t)(unsigned)(size_t)(void*)l, 0, 0);
#else
  unsigned ldsOff = (unsigned)(size_t)(void*)l;  // low 32 bits = LDS offset
  asm volatile("global_load_async_to_lds_b128 %0, %1, off"
               :: "v"(ldsOff), "v"(g) : "memory");
#endif
}

__device__ __forceinline__ void wait_async0() {
#if __has_builtin(__builtin_amdgcn_s_wait_asynccnt)
  __builtin_amdgcn_s_wait_asynccnt(0);
#else
  asm volatile("s_wait_asynccnt 0x0" ::: "memory");
#endif
}

// ---------------------------------------------------------------------------
// Kernel 1a: split-D fused WMMA distance tiles + running argmin.
// Block = 8 waves. blockIdx.x = D-slice, blockIdx.y = group of 8 query-tiles.
// The block async-stages its 4096-point slice into LDS once; each wave owns a
// 16-query tile and streams 256 16-column tiles through
// V_WMMA_F32_16X16X4_F32, keeping per-row (min, argmin) in registers.
// Writes per-slice partials to the workspace.
// ---------------------------------------------------------------------------
__global__ __launch_bounds__(256) void hard_nn_slice(
    const float* __restrict__ query,   // [B,2]
    const float* __restrict__ db,      // [D,2]
    float* __restrict__ wsv,           // [S,B] partial min sq-dist (shifted by -||q||^2)
    int*   __restrict__ wsi,           // [S,B] partial argmin
    int B, int D)
{
  __shared__ __align__(16) float2 tile[SLICE_POINTS];
  float* tf = (float*)tile;

  const int slice      = blockIdx.x;
  const int sliceStart = slice * SLICE_POINTS;
  int validPts = D - sliceStart;
  if (validPts > SLICE_POINTS) validPts = SLICE_POINTS;

  // Pad tail with huge coords -> huge distance, never selected.
  for (int i = validPts * 2 + threadIdx.x; i < SLICE_POINTS * 2; i += blockDim.x)
    tf[i] = 1.0e18f;

  // Async-stage the valid region in 16B chunks (ASYNCcnt), tail via plain copy.
  const int chunks = (validPts * 2) >> 2;               // float4 chunks
  for (int c = threadIdx.x; c < chunks; c += blockDim.x)
    async_copy_b128(db + (size_t)sliceStart * 2 + c * 4, tf + c * 4);
  for (int i = chunks * 4 + threadIdx.x; i < validPts * 2; i += blockDim.x)
    tf[i] = db[(size_t)sliceStart * 2 + i];

  wait_async0();
  __syncthreads();

  const int lane  = threadIdx.x & 31;
  const int wave  = threadIdx.x >> 5;
  const int qtile = blockIdx.y * 8 + wave;
  const int numQTiles = (B + 15) >> 4;
  if (qtile < numQTiles) {                 // wave-uniform -> EXEC all-1s at WMMA
    const int qbase = qtile << 4;
    const int half  = lane >> 4;
    const int nlane = lane & 15;

    // A operand: 16x4 rows = queries. lanes 0-15: (qx,qy)=K0,K1; upper: K2,K3=0.
    int arow = qbase + nlane; if (arow >= B) arow = B - 1;
    v2f a;
    if (half == 0) { a.x = query[2 * arow]; a.y = query[2 * arow + 1]; }
    else           { a.x = 0.0f;            a.y = 0.0f; }

    float best[8];
    int   bidx[8];
#pragma unroll
    for (int r = 0; r < 8; ++r) { best[r] = 3.4e38f; bidx[r] = 0; }

    const float2* tp = (const float2*)tile;
    for (int cb = 0; cb < SLICE_POINTS; cb += 16) {
      const float2 p = tp[cb + nlane];
      const float pn = fmaf(p.x, p.x, p.y * p.y);
      v2f b;
      b.x = half ? 0.0f : p.x;
      b.y = half ? 0.0f : p.y;

      v8f acc = {};
      acc = __builtin_amdgcn_wmma_f32_16x16x4_f32(
          false, a, false, b, (short)0, acc, false, false);

      const int col = sliceStart + cb + nlane;
      // argmin of ||p||^2 - 2<q,p>  ==  argmin of sq-dist (||q||^2 is a
      // per-row constant). Strict < on ascending cols == first-occurrence.
#pragma unroll
      for (int r = 0; r < 8; ++r) {
        const float dval = fmaf(-2.0f, acc[r], pn);
        if (dval < best[r]) { best[r] = dval; bidx[r] = col; }
      }
    }

    // Reduce (min, argmin) across the 16 lanes of each half-wave.
#pragma unroll
    for (int m = 1; m <= 8; m <<= 1) {
#pragma unroll
      for (int r = 0; r < 8; ++r) {
        const float ov = __shfl_xor(best[r], m, 32);
        const int   oi = __shfl_xor(bidx[r], m, 32);
        if (ov < best[r] || (ov == best[r] && oi < bidx[r])) {
          best[r] = ov; bidx[r] = oi;
        }
      }
    }

    if (nlane == 0) {
#pragma unroll
      for (int r = 0; r < 8; ++r) {
        const int m = qbase + 8 * half + r;
        if (m < B) {
          wsv[(size_t)slice * B + m] = best[r];
          wsi[(size_t)slice * B + m] = bidx[r];
        }
      }
    }
  }
}

// ---------------------------------------------------------------------------
// Kernel 1b: fold per-slice partials, gather metric.
// ---------------------------------------------------------------------------
__global__ __launch_bounds__(256) void hard_reduce(
    const float* __restrict__ wsv, const int* __restrict__ wsi,
    const float* __restrict__ metrics, float* __restrict__ out_hard,
    int B, int S)
{
  const int m = blockIdx.x * blockDim.x + threadIdx.x;
  if (m >= B) return;
  float bv = 3.4e38f;
  int   bi = 0x7fffffff;
  for (int s = 0; s < S; ++s) {              // ascending slices: idx ranges ascend
    const float v = wsv[(size_t)s * B + m];
    const int   i = wsi[(size_t)s * B + m];
    if (v < bv || (v == bv && i < bi)) { bv = v; bi = i; }
  }
  out_hard[m] = metrics[bi];
}

// ---------------------------------------------------------------------------
// Fallback (workspace too small): monolithic wave-per-qtile WMMA argmin.
// ---------------------------------------------------------------------------
__global__ __launch_bounds__(256) void hard_nn_mono(
    const float* __restrict__ query, const float* __restrict__ db,
    const float* __restrict__ metrics, float* __restrict__ out_hard,
    int B, int D)
{
  const int lane  = threadIdx.x & 31;
  const int qtile = blockIdx.x * (blockDim.x >> 5) + (threadIdx.x >> 5);
  if (qtile >= (B >> 4)) return;
  const int qbase = qtile << 4;
  const int half  = lane >> 4;
  const int nlane = lane & 15;

  v2f a;
  if (half == 0) { a.x = query[2 * (qbase + nlane)]; a.y = query[2 * (qbase + nlane) + 1]; }
  else           { a.x = 0.0f; a.y = 0.0f; }

  float best[8]; int bidx[8];
#pragma unroll
  for (int r = 0; r < 8; ++r) { best[r] = 3.4e38f; bidx[r] = 0; }

  for (int cb = 0; cb < D; cb += 16) {
    const int col = cb + nlane;
    const float px = db[2 * col], py = db[2 * col + 1];
    const float pn = fmaf(px, px, py * py);
    v2f b; b.x = half ? 0.0f : px; b.y = half ? 0.0f : py;
    v8f acc = {};
    acc = __builtin_amdgcn_wmma_f32_16x16x4_f32(
        false, a, false, b, (short)0, acc, false, false);
#pragma unroll
    for (int r = 0; r < 8; ++r) {
      const float dval = fmaf(-2.0f, acc[r], pn);
      if (dval < best[r]) { best[r] = dval; bidx[r] = col; }
    }
  }
#pragma unroll
  for (int m = 1; m <= 8; m <<= 1) {
#pragma unroll
    for (int r = 0; r < 8; ++r) {
      const float ov = __shfl_xor(best[r], m, 32);
      const int   oi = __shfl_xor(bidx[r], m, 32);
      if (ov < best[r] || (ov == best[r] && oi < bidx[r])) { best[r] = ov; bidx[r] = oi; }
    }
  }
  if (nlane == 0) {
#pragma unroll
    for (int r = 0; r < 8; ++r)
      out_hard[qbase + 8 * half + r] = metrics[bidx[r]];
  }
}

// ---------------------------------------------------------------------------
// Kernel 2: windowed softmax gather. One wave per query; wave32 shuffles.
// ---------------------------------------------------------------------------
__global__ __launch_bounds__(256) void soft_local_kernel(
    const float* __restrict__ query, const float* __restrict__ temp,
    const float* __restrict__ grid2d, const float* __restrict__ met2d,
    const float* __restrict__ spacing, const int* __restrict__ ksz,
    float* __restrict__ out_soft, int B, int NB, int NS)
{
  const int lane = threadIdx.x & 31;
  const int q = blockIdx.x * (blockDim.x >> 5) + (threadIdx.x >> 5);
  if (q >= B) return;

  const float T  = temp[0] + 1e-8f;
  const float ox = grid2d[0], oy = grid2d[1];
  const float sx = spacing[0], sy = spacing[1];
  const int radius = ksz[0] >> 1;
  const int kw = 2 * radius + 1;
  const int K2 = kw * kw;

  const float qx = query[2 * q], qy = query[2 * q + 1];
  const int vx = (int)rintf((qx - ox) / sx);   // jnp.round == half-to-even
  const int vy = (int)rintf((qy - oy) / sy);

  float mx = -3.4e38f;
  for (int k = lane; k < K2; k += 32) {
    int x = vx + (k / kw - radius);
    int y = vy + (k % kw - radius);
    x = x < 0 ? 0 : (x >= NB ? NB - 1 : x);    // JAX clips OOB gathers
    y = y < 0 ? 0 : (y >= NS ? NS - 1 : y);
    const int idx = x * NS + y;
    const float dx = qx - grid2d[2 * idx], dy = qy - grid2d[2 * idx + 1];
    mx = fmaxf(mx, -(dx * dx + dy * dy) / T);
  }
#pragma unroll
  for (int m = 1; m < 32; m <<= 1) mx = fmaxf(mx, __shfl_xor(mx, m, 32));

  float se = 0.0f, ws = 0.0f;
  for (int k = lane; k < K2; k += 32) {
    int x = vx + (k / kw - radius);
    int y = vy + (k % kw - radius);
    x = x < 0 ? 0 : (x >= NB ? NB - 1 : x);
    y = y < 0 ? 0 : (y >= NS ? NS - 1 : y);
    const int idx = x * NS + y;
    const float dx = qx - grid2d[2 * idx], dy = qy - grid2d[2 * idx + 1];
    const float e = expf(-(dx * dx + dy * dy) / T - mx);
    se += e;
    ws += e * met2d[idx];
  }
#pragma unroll
  for (int m = 1; m < 32; m <<= 1) {
    se += __shfl_xor(se, m, 32);
    ws += __shfl_xor(ws, m, 32);
  }
  if (lane == 0) out_soft[q] = ws / se;
}

// ---------------------------------------------------------------------------
extern "C" void kernel_launch(void* const* d_in, const int* in_sizes, int n_in,
                              void* d_out, int out_size, void* d_ws, size_t ws_size,
                              hipStream_t stream) {
  const float* query   = (const float*)d_in[0];
  const float* temp    = (const float*)d_in[1];
  const float* db      = (const float*)d_in[2];
  const float* metrics = (const float*)d_in[3];
  const float* grid2d  = (const float*)d_in[4];
  const float* met2d   = (const float*)d_in[5];
  const float* spacing = (const float*)d_in[6];
  const int*   ksz     = (const int*)d_in[7];
  float* out = (float*)d_out;

  const int B = in_sizes[0] / 2;
  const int D = in_sizes[3];
  int ns = 1;                       // grid is square (NB == NS): integer sqrt
  while ((ns + 1) * (ns + 1) <= D) ++ns;
  const int nb = D / ns;

  const dim3 blk(256);

  const int S = (D + SLICE_POINTS - 1) / SLICE_POINTS;
  const size_t need = (size_t)S * B * (sizeof(float) + sizeof(int));
  if (ws_size >= need) {
    float* wsv = (float*)d_ws;
    int*   wsi = (int*)(wsv + (size_t)S * B);
    const int qtiles  = (B + 15) / 16;
    const int qgroups = (qtiles + 7) / 8;
    hipLaunchKernelGGL(hard_nn_slice, dim3(S, qgroups), blk, 0, stream,
                       query, db, wsv, wsi, B, D);
    hipLaunchKernelGGL(hard_reduce, dim3((B + 255) / 256), blk, 0, stream,
                       wsv, wsi, metrics, out, B, S);
  } else {
    const int qtiles = B / 16;
    hipLaunchKernelGGL(hard_nn_mono, dim3((qtiles + 7) / 8), blk, 0, stream,
                       query, db, metrics, out, B, D);
  }

  const int softBlocks = (B + 7) / 8;
  hipLaunchKernelGGL(soft_local_kernel, dim3(softBlocks), blk, 0, stream,
                     query, temp, grid2d, met2d, spacing, ksz, out + B, B, nb, ns);
}